// StructuralLoss_36524401885682
// MI455X (gfx1250) — compile-verified
//
#include <hip/hip_runtime.h>

typedef __attribute__((ext_vector_type(2))) float v2f;
typedef __attribute__((ext_vector_type(8))) float v8f;

// ---------------------------------------------------------------------------
// Wave32 all-lane sum using the f32 WMMA unit (no precision loss: f32 in/out).
// A (16x4): VGPR0 = per-lane partial  -> A[m][0] = p_m (lanes 0-15),
//                                        A[m][2] = p_(m+16) (lanes 16-31)
//           VGPR1 = 0                 -> A[m][1] = A[m][3] = 0
// B (4x16): all ones.
// D[m][n] = p_m + p_(m+16) for every n.  Summing a lane's 8 D-VGPRs gives the
// half-wave total for its lane group; one xor-16 shuffle combines the halves.
// ---------------------------------------------------------------------------
__device__ __forceinline__ float wave_sum32(float p) {
#if __has_builtin(__builtin_amdgcn_wmma_f32_16x16x4_f32)
    v2f a; a[0] = p;   a[1] = 0.0f;
    v2f b; b[0] = 1.0f; b[1] = 1.0f;
    v8f c = {};
    c = __builtin_amdgcn_wmma_f32_16x16x4_f32(
        /*neg_a=*/false, a, /*neg_b=*/false, b,
        /*c_mod=*/(short)0, c, /*reuse_a=*/false, /*reuse_b=*/false);
    float t = ((c[0] + c[1]) + (c[2] + c[3])) + ((c[4] + c[5]) + (c[6] + c[7]));
    t += __shfl_xor(t, 16, 32);
    return t;
#else
#warning "wmma_f32_16x16x4_f32 builtin unavailable on this toolchain; using shuffle fallback"
    for (int o = 16; o > 0; o >>= 1) p += __shfl_xor(p, o, 32);
    return p;
#endif
}

__global__ void zero_out_kernel(float* out) { out[0] = 0.0f; }

__global__ __launch_bounds__(256)
void structural_loss_kernel(const float* __restrict__ pred, float* __restrict__ out,
                            long long rows, float scale) {
    const long long nth = (long long)gridDim.x * blockDim.x;
    float acc = 0.0f;

    for (long long row = (long long)blockIdx.x * blockDim.x + threadIdx.x;
         row < rows; row += nth) {
        // Prefetch next grid-stride row (global_prefetch_b8); cheap no-op at tail.
        if (row + nth < rows)
            __builtin_prefetch(pred + (row + nth) * 60LL, 0, 1);

        // One row = 60 f32 = 15 aligned float4 -> global_load_b128 x15.
        const float4* __restrict__ rp =
            reinterpret_cast<const float4*>(pred + row * 60LL);
        float p[60];
#pragma unroll
        for (int j = 0; j < 15; ++j) {
            float4 v = rp[j];
            p[4 * j + 0] = v.x; p[4 * j + 1] = v.y;
            p[4 * j + 2] = v.z; p[4 * j + 3] = v.w;
        }

        // 3 keys; per key: bid[l]=p[20k+2l], ask[l]=p[20k+10+2l], l=0..4
        float s = 0.0f;
#pragma unroll
        for (int k = 0; k < 3; ++k) {
            const int b0 = 20 * k;
            const int a0 = 20 * k + 10;
#pragma unroll
            for (int l = 0; l < 4; ++l) {
                float av = p[a0 + 2 * l]     - p[a0 + 2 * l + 2]; // ask[l]-ask[l+1]
                float bv = p[b0 + 2 * l + 2] - p[b0 + 2 * l];     // bid[l+1]-bid[l]
                s += fmaxf(av, 0.0f) + fmaxf(bv, 0.0f);
            }
            s += fmaxf(p[b0] - p[a0], 0.0f);                      // bid[0]-ask[0]
        }
        acc += s;
    }

    // Wave reduction via f32 WMMA (all 32 lanes active here: loop guard has
    // reconverged, EXEC is all ones as WMMA requires).
    float wtot = wave_sum32(acc);

    // Block reduction: 8 wave32s -> LDS -> one f32 atomic per block.
    __shared__ float wsum[8];
    const int wave = threadIdx.x >> 5;
    if ((threadIdx.x & 31) == 0) wsum[wave] = wtot;
    __syncthreads();
    if (threadIdx.x == 0) {
        float b = ((wsum[0] + wsum[1]) + (wsum[2] + wsum[3])) +
                  ((wsum[4] + wsum[5]) + (wsum[6] + wsum[7]));
        atomicAdd(out, b * scale);
    }
}

extern "C" void kernel_launch(void* const* d_in, const int* in_sizes, int n_in,
                              void* d_out, int out_size, void* d_ws, size_t ws_size,
                              hipStream_t stream) {
    (void)n_in; (void)out_size; (void)d_ws; (void)ws_size;
    const float* pred = (const float*)d_in[0];
    float* out = (float*)d_out;

    const long long rows = (long long)in_sizes[0] / 60;   // 64*8192 = 524288
    const float scale = 1.0f / 64.0f;                     // mean over batch axis

    zero_out_kernel<<<1, 1, 0, stream>>>(out);

    int blocks = (int)((rows + 255) / 256);
    if (blocks > 1024) blocks = 1024;                     // 2 rows/thread, 8192 waves
    if (blocks < 1) blocks = 1;
    structural_loss_kernel<<<blocks, 256, 0, stream>>>(pred, out, rows, scale);
}